// collisioner_23854248362460
// MI455X (gfx1250) — compile-verified
//
#include <hip/hip_runtime.h>
#include <math.h>
#include <utility>

#define NN 48
#define VOL (NN * NN * NN)   // 110592
#define BATCH 8
#define TMODES 20            // (M-1)*M with M=5
#define NPAIR (NN / 2)       // 24 K-pairs per column
#define LDP 52               // padded LDS row stride (multiple of 4 floats)

typedef float v2f __attribute__((ext_vector_type(2)));
typedef float v8f __attribute__((ext_vector_type(8)));

// ---------------------------------------------------------------------------
// Twiddle init, pre-paired for WMMA B-fragments:
//   Wp[(col*24 + j)*2 + {0,1}]        = cos(-2pi*(2j)*col/48), cos(-2pi*(2j+1)*col/48)
//   Wp[2304 + (col*24 + j)*2 + {0,1}] = sin(...), sin(...)
// ---------------------------------------------------------------------------
__global__ void coll_winit_kernel(float* __restrict__ W) {
    int idx = blockIdx.x * blockDim.x + threadIdx.x;   // over 48*24
    if (idx >= NN * NPAIR) return;
    int col = idx / NPAIR, j = idx % NPAIR;
    int x0 = 2 * j, x1 = 2 * j + 1;
    const float w0 = -2.0f * 3.14159265358979323846f / (float)NN;
    float s0, c0, s1, c1;
    __sincosf(w0 * (float)((x0 * col) % NN), &s0, &c0);
    __sincosf(w0 * (float)((x1 * col) % NN), &s1, &c1);
    W[idx * 2]     = c0;
    W[idx * 2 + 1] = c1;
    W[NN * NPAIR * 2 + idx * 2]     = s0;
    W[NN * NPAIR * 2 + idx * 2 + 1] = s1;
}

// ---------------------------------------------------------------------------
// fftshift-folded prep (float4): u = f * (-1)^(x+y+z) / N^3, sign alternates
// {+,-,+,-} within each x-quad.
// ---------------------------------------------------------------------------
__global__ void coll_prep_kernel(const float* __restrict__ f,
                                 float* __restrict__ ur, float* __restrict__ ui) {
    int idx = blockIdx.x * blockDim.x + threadIdx.x;
    if (idx >= (BATCH * VOL) / 4) return;
    int v = (idx * 4) % VOL;
    int x = v % NN, y = (v / NN) % NN, z = v / (NN * NN);
    float s = ((x + y + z) & 1) ? -1.0f : 1.0f;
    s *= (1.0f / (float)VOL);
    float4 fv = ((const float4*)f)[idx];
    float4 o; o.x = fv.x * s; o.y = -fv.y * s; o.z = fv.z * s; o.w = -fv.w * s;
    ((float4*)ur)[idx] = o;
    float4 zz; zz.x = 0.f; zz.y = 0.f; zz.z = 0.f; zz.w = 0.f;
    ((float4*)ui)[idx] = zz;
}

// ---------------------------------------------------------------------------
// 1D DFT along one axis as WMMA GEMM (f32 16x16x4).
// Block = 96 threads = 3 waves; each wave owns TWO 16-line tiles (32 lines)
// sharing B fragments. grid.x = 24, grid.y = batch.
// mode 0 (axis x): addr = r*48 + col          (tile is fully contiguous)
// mode 1 (axis y): addr = (r/48)*2304 + r%48 + col*48
// mode 2 (axis z): addr = r + col*2304
// All mode branches are SGPR-uniform; EXEC stays all-ones for WMMA.
// ---------------------------------------------------------------------------
__global__ __launch_bounds__(96)
void coll_fft_axis_kernel(const float* __restrict__ sr, const float* __restrict__ si,
                          float* __restrict__ dr, float* __restrict__ di,
                          const float* __restrict__ Wp,
                          int mode) {
    __shared__ float tr[3][32][LDP];
    __shared__ float ti[3][32][LDP];
    __shared__ float wrp[NN * NPAIR * 2];   // paired cos
    __shared__ float wip[NN * NPAIR * 2];   // paired sin

    const int tid  = threadIdx.x;
    const int w    = tid / 32;
    const int lane = tid % 32;
    const size_t vbase = (size_t)blockIdx.y * VOL;

    for (int i = tid; i < NN * NPAIR * 2; i += 96) {
        wrp[i] = Wp[i];
        wip[i] = Wp[NN * NPAIR * 2 + i];
    }

    const int r0 = (blockIdx.x * 3 + w) * 32;
    const int a0 = r0 / 48;          // only used by mode 1 (one-time)
    const int b0 = r0 - a0 * 48;     // in {0,16,32}
    float* trw = &tr[w][0][0];
    float* tiw = &ti[w][0][0];

    if (mode == 0) {
        // Whole wave tile is contiguous [r0*48, r0*48+1536). 384 float4 quads.
        int row = lane / 12;              // one-time tiny division
        int c   = lane - row * 12;
        int loff = row * LDP + c * 4;     // LDS float offset, 16B aligned
        size_t g = vbase + (size_t)r0 * 48 + (size_t)lane * 4;
#pragma unroll
        for (int it = 0; it < 12; ++it) {
            float4 vr = *(const float4*)(sr + g);
            float4 vi = *(const float4*)(si + g);
            *(float4*)(trw + loff) = vr;
            *(float4*)(tiw + loff) = vi;
            g += 128;                     // 32 quads
            c += 8;
            int ov = (c >= 12) ? 1 : 0;   // branch-free carry
            c -= 12 * ov;
            loff += 2 * LDP + 32 + ov * (LDP - 48);   // +136 (+4 on carry)
        }
    } else {
        // Lines contiguous in memory: float4 along rows, scatter into LDS.
        const int col0 = lane >> 3;       // 0..3
        const int rowb = (lane & 7) * 4;  // 0,4,..,28 ; rowb..rowb+3 never cross 48
        size_t gbase, gstep;
        if (mode == 1) {
            int t  = b0 + rowb;
            int ov = (t >= 48) ? 1 : 0;
            gbase = vbase + (size_t)(a0 + ov) * 2304 + (size_t)(t - 48 * ov) + (size_t)col0 * 48;
            gstep = 4 * 48;
        } else {
            gbase = vbase + (size_t)(r0 + rowb) + (size_t)col0 * 2304;
            gstep = (size_t)4 * 2304;
        }
        int lo = rowb * LDP + col0;       // LDS float offset of [rowb][col0]
#pragma unroll
        for (int it = 0; it < 12; ++it) {
            float4 vr = *(const float4*)(sr + gbase);
            float4 vi = *(const float4*)(si + gbase);
            trw[lo]           = vr.x; trw[lo + LDP]     = vr.y;
            trw[lo + 2 * LDP] = vr.z; trw[lo + 3 * LDP] = vr.w;
            tiw[lo]           = vi.x; tiw[lo + LDP]     = vi.y;
            tiw[lo + 2 * LDP] = vi.z; tiw[lo + 3 * LDP] = vi.w;
            gbase += gstep;
            lo += 4;                      // col += 4
        }
    }
    __syncthreads();

    // f32 WMMA 16x16x4 fragment mapping (ISA 7.12.2).
    const int m    = lane & 15;
    const int half = lane >> 4;          // 0 or 1
    const int moff = half << 3;          // 0 or 8

    for (int kt = 0; kt < 3; ++kt) {
        v8f cr0  = {0.f,0.f,0.f,0.f,0.f,0.f,0.f,0.f};
        v8f crn0 = {0.f,0.f,0.f,0.f,0.f,0.f,0.f,0.f};
        v8f ci0  = {0.f,0.f,0.f,0.f,0.f,0.f,0.f,0.f};
        v8f cr1  = {0.f,0.f,0.f,0.f,0.f,0.f,0.f,0.f};
        v8f crn1 = {0.f,0.f,0.f,0.f,0.f,0.f,0.f,0.f};
        v8f ci1  = {0.f,0.f,0.f,0.f,0.f,0.f,0.f,0.f};
        const int col = kt * 16 + m;
#pragma unroll
        for (int k0 = 0; k0 < NN; k0 += 4) {
            const int jp = (k0 >> 1) + half;       // twiddle pair index
            const int ka = k0 + (half << 1);       // A K-offset (even)
            v2f br  = *(const v2f*)&wrp[(col * NPAIR + jp) * 2];
            v2f bi  = *(const v2f*)&wip[(col * NPAIR + jp) * 2];
            v2f a0r = *(const v2f*)(trw + m * LDP + ka);
            v2f a0i = *(const v2f*)(tiw + m * LDP + ka);
            v2f a1r = *(const v2f*)(trw + (16 + m) * LDP + ka);
            v2f a1i = *(const v2f*)(tiw + (16 + m) * LDP + ka);
            cr0  = __builtin_amdgcn_wmma_f32_16x16x4_f32(false, a0r, false, br, (short)0, cr0,  false, false);
            crn0 = __builtin_amdgcn_wmma_f32_16x16x4_f32(false, a0i, false, bi, (short)0, crn0, false, false);
            ci0  = __builtin_amdgcn_wmma_f32_16x16x4_f32(false, a0r, false, bi, (short)0, ci0,  false, false);
            ci0  = __builtin_amdgcn_wmma_f32_16x16x4_f32(false, a0i, false, br, (short)0, ci0,  false, false);
            cr1  = __builtin_amdgcn_wmma_f32_16x16x4_f32(false, a1r, false, br, (short)0, cr1,  false, false);
            crn1 = __builtin_amdgcn_wmma_f32_16x16x4_f32(false, a1i, false, bi, (short)0, crn1, false, false);
            ci1  = __builtin_amdgcn_wmma_f32_16x16x4_f32(false, a1r, false, bi, (short)0, ci1,  false, false);
            ci1  = __builtin_amdgcn_wmma_f32_16x16x4_f32(false, a1i, false, br, (short)0, ci1,  false, false);
        }

        if (mode == 0) {
            // g = r*48 + col ; per row: g += 48. Lanes coalesce along col.
            size_t g0 = vbase + (size_t)(r0 + moff) * 48 + (size_t)col;
            size_t g1 = vbase + (size_t)(r0 + 16 + moff) * 48 + (size_t)col;
#pragma unroll
            for (int v = 0; v < 8; ++v) {
                dr[g0] = cr0[v] - crn0[v]; di[g0] = ci0[v]; g0 += 48;
                dr[g1] = cr1[v] - crn1[v]; di[g1] = ci1[v]; g1 += 48;
            }
        } else {
            // Each lane's 8 output rows are contiguous: two float4 stores/plane/half.
            size_t g0, g1;
            if (mode == 1) {
                int t0 = b0 + moff;      int ov0 = (t0 >= 48) ? 1 : 0;
                int t1 = b0 + 16 + moff; int ov1 = (t1 >= 48) ? 1 : 0;
                g0 = vbase + (size_t)(a0 + ov0) * 2304 + (size_t)(t0 - 48 * ov0) + (size_t)col * 48;
                g1 = vbase + (size_t)(a0 + ov1) * 2304 + (size_t)(t1 - 48 * ov1) + (size_t)col * 48;
            } else {
                g0 = vbase + (size_t)(r0 + moff) + (size_t)col * 2304;
                g1 = vbase + (size_t)(r0 + 16 + moff) + (size_t)col * 2304;
            }
            float4 o0, o1;
            o0.x = cr0[0]-crn0[0]; o0.y = cr0[1]-crn0[1]; o0.z = cr0[2]-crn0[2]; o0.w = cr0[3]-crn0[3];
            o1.x = cr0[4]-crn0[4]; o1.y = cr0[5]-crn0[5]; o1.z = cr0[6]-crn0[6]; o1.w = cr0[7]-crn0[7];
            *(float4*)(dr + g0) = o0; *(float4*)(dr + g0 + 4) = o1;
            o0.x = ci0[0]; o0.y = ci0[1]; o0.z = ci0[2]; o0.w = ci0[3];
            o1.x = ci0[4]; o1.y = ci0[5]; o1.z = ci0[6]; o1.w = ci0[7];
            *(float4*)(di + g0) = o0; *(float4*)(di + g0 + 4) = o1;
            o0.x = cr1[0]-crn1[0]; o0.y = cr1[1]-crn1[1]; o0.z = cr1[2]-crn1[2]; o0.w = cr1[3]-crn1[3];
            o1.x = cr1[4]-crn1[4]; o1.y = cr1[5]-crn1[5]; o1.z = cr1[6]-crn1[6]; o1.w = cr1[7]-crn1[7];
            *(float4*)(dr + g1) = o0; *(float4*)(dr + g1 + 4) = o1;
            o0.x = ci1[0]; o0.y = ci1[1]; o0.z = ci1[2]; o0.w = ci1[3];
            o1.x = ci1[4]; o1.y = ci1[5]; o1.z = ci1[6]; o1.w = ci1[7];
            *(float4*)(di + g1) = o0; *(float4*)(di + g1 + 4) = o1;
        }
    }
}

// ---------------------------------------------------------------------------
// Elementwise helpers (float4 streams)
// ---------------------------------------------------------------------------
__global__ void coll_mul2_kernel(const float* __restrict__ fr, const float* __restrict__ fi,
                                 const float* __restrict__ ph, const float* __restrict__ ps,
                                 float* __restrict__ ar, float* __restrict__ ai,
                                 float* __restrict__ br, float* __restrict__ bi) {
    int idx = blockIdx.x * blockDim.x + threadIdx.x;
    if (idx >= (BATCH * VOL) / 4) return;
    int vv = idx % (VOL / 4);
    float4 p = ((const float4*)ph)[vv];
    float4 q = ((const float4*)ps)[vv];
    float4 xr = ((const float4*)fr)[idx];
    float4 xi = ((const float4*)fi)[idx];
    float4 o;
    o.x = xr.x*p.x; o.y = xr.y*p.y; o.z = xr.z*p.z; o.w = xr.w*p.w; ((float4*)ar)[idx] = o;
    o.x = xi.x*p.x; o.y = xi.y*p.y; o.z = xi.z*p.z; o.w = xi.w*p.w; ((float4*)ai)[idx] = o;
    o.x = xr.x*q.x; o.y = xr.y*q.y; o.z = xr.z*q.z; o.w = xr.w*q.w; ((float4*)br)[idx] = o;
    o.x = xi.x*q.x; o.y = xi.y*q.y; o.z = xi.z*q.z; o.w = xi.w*q.w; ((float4*)bi)[idx] = o;
}

// ar,ai = fspec*phipsi ; br,bi = copy of fspec
__global__ void coll_mul1_kernel(const float* __restrict__ fr, const float* __restrict__ fi,
                                 const float* __restrict__ pp,
                                 float* __restrict__ ar, float* __restrict__ ai,
                                 float* __restrict__ br, float* __restrict__ bi) {
    int idx = blockIdx.x * blockDim.x + threadIdx.x;
    if (idx >= (BATCH * VOL) / 4) return;
    int vv = idx % (VOL / 4);
    float4 p = ((const float4*)pp)[vv];
    float4 xr = ((const float4*)fr)[idx];
    float4 xi = ((const float4*)fi)[idx];
    float4 o;
    o.x = xr.x*p.x; o.y = xr.y*p.y; o.z = xr.z*p.z; o.w = xr.w*p.w; ((float4*)ar)[idx] = o;
    o.x = xi.x*p.x; o.y = xi.y*p.y; o.z = xi.z*p.z; o.w = xi.w*p.w; ((float4*)ai)[idx] = o;
    ((float4*)br)[idx] = xr;
    ((float4*)bi)[idx] = xi;
}

// fg(real) accumulation: only Re(A*B) is needed downstream.
__global__ void coll_acc_kernel(float* __restrict__ fg,
                                const float* __restrict__ ar, const float* __restrict__ ai,
                                const float* __restrict__ br, const float* __restrict__ bi,
                                int first) {
    int idx = blockIdx.x * blockDim.x + threadIdx.x;
    if (idx >= (BATCH * VOL) / 4) return;
    float4 xr = ((const float4*)ar)[idx];
    float4 xi = ((const float4*)ai)[idx];
    float4 yr = ((const float4*)br)[idx];
    float4 yi = ((const float4*)bi)[idx];
    float4 re;
    re.x = xr.x*yr.x - xi.x*yi.x;
    re.y = xr.y*yr.y - xi.y*yi.y;
    re.z = xr.z*yr.z - xi.z*yi.z;
    re.w = xr.w*yr.w - xi.w*yi.w;
    if (!first) {
        float4 g = ((const float4*)fg)[idx];
        re.x += g.x; re.y += g.y; re.z += g.z; re.w += g.w;
    }
    ((float4*)fg)[idx] = re;
}

__global__ void coll_final_kernel(const float* __restrict__ fg,
                                  const float* __restrict__ g1r, const float* __restrict__ g1i,
                                  const float* __restrict__ g2r, const float* __restrict__ g2i,
                                  const float* __restrict__ kn, const int* __restrict__ Mp,
                                  float* __restrict__ out) {
    int idx = blockIdx.x * blockDim.x + threadIdx.x;
    if (idx >= (BATCH * VOL) / 4) return;
    float M = (float)Mp[0];
    float scale = 4.0f * 9.869604401089358f / (kn[0] * M * M);   // 4*pi^2 / kn / M^2
    float4 xr = ((const float4*)g1r)[idx];
    float4 xi = ((const float4*)g1i)[idx];
    float4 yr = ((const float4*)g2r)[idx];
    float4 yi = ((const float4*)g2i)[idx];
    float4 g  = ((const float4*)fg)[idx];
    float4 o;
    o.x = scale * (g.x - (xr.x*yr.x - xi.x*yi.x));
    o.y = scale * (g.y - (xr.y*yr.y - xi.y*yi.y));
    o.z = scale * (g.z - (xr.z*yr.z - xi.z*yi.z));
    o.w = scale * (g.w - (xr.w*yr.w - xi.w*yi.w));
    ((float4*)out)[idx] = o;
}

// ---------------------------------------------------------------------------
// Host orchestration (all on `stream`, graph-capture safe)
// ---------------------------------------------------------------------------
extern "C" void kernel_launch(void* const* d_in, const int* in_sizes, int n_in,
                              void* d_out, int out_size, void* d_ws, size_t ws_size,
                              hipStream_t stream) {
    const float* f   = (const float*)d_in[0];
    const float* kn  = (const float*)d_in[1];
    const float* phi = (const float*)d_in[2];
    const float* psi = (const float*)d_in[3];
    const float* pp  = (const float*)d_in[4];
    const int*   Mp  = (const int*)d_in[5];
    float* out = (float*)d_out;

    float* ws = (float*)d_ws;
    const size_t S = (size_t)BATCH * VOL;   // one plane: 884736 floats

    float* Wm = ws;                 // paired twiddles (2 * 2304 floats)
    float* P  = ws + 8192;          // 11 planes of S floats each (~39 MB total)
    float* fr   = P + 0 * S; float* fi   = P + 1 * S;   // f_spec
    float* ar_  = P + 2 * S; float* ai_  = P + 3 * S;   // stream A
    float* asr  = P + 4 * S; float* asi  = P + 5 * S;   // scratch for A
    float* br_  = P + 6 * S; float* bi_  = P + 7 * S;   // stream B
    float* bsr  = P + 8 * S; float* bsi  = P + 9 * S;   // scratch for B
    float* fg   = P + 10 * S;                           // real accumulator

    const int EB4 = (int)((S / 4 + 255) / 256);

    coll_winit_kernel<<<dim3((NN * NPAIR + 255) / 256), 256, 0, stream>>>(Wm);
    coll_prep_kernel<<<EB4, 256, 0, stream>>>(f, fr, fi);

    auto fftax = [&](const float* xr, const float* xi, float* yr, float* yi, int mode) {
        coll_fft_axis_kernel<<<dim3(24, BATCH), 96, 0, stream>>>(xr, xi, yr, yi, Wm, mode);
    };
    auto fft3 = [&](float*& xr, float*& xi, float*& sr2, float*& si2) {
        fftax(xr, xi, sr2, si2, 0); std::swap(xr, sr2); std::swap(xi, si2);
        fftax(xr, xi, sr2, si2, 1); std::swap(xr, sr2); std::swap(xi, si2);
        fftax(xr, xi, sr2, si2, 2); std::swap(xr, sr2); std::swap(xi, si2);
    };

    // f_spec = fftshift(FFT3(f))/N^3  (shift folded into prep sign flip)
    fft3(fr, fi, asr, asi);

    // Gain term: sum_t Re( FFT3(fspec*phi_t) * FFT3(fspec*psi_t) )
    for (int t = 0; t < TMODES; ++t) {
        coll_mul2_kernel<<<EB4, 256, 0, stream>>>(fr, fi,
                                                  phi + (size_t)t * VOL, psi + (size_t)t * VOL,
                                                  ar_, ai_, br_, bi_);
        fft3(ar_, ai_, asr, asi);
        fft3(br_, bi_, bsr, bsi);
        coll_acc_kernel<<<EB4, 256, 0, stream>>>(fg, ar_, ai_, br_, bi_, t == 0 ? 1 : 0);
    }

    // Loss term: Re( FFT3(fspec*phipsi) * FFT3(fspec) )
    coll_mul1_kernel<<<EB4, 256, 0, stream>>>(fr, fi, pp, ar_, ai_, br_, bi_);
    fft3(ar_, ai_, asr, asi);
    fft3(br_, bi_, bsr, bsi);

    coll_final_kernel<<<EB4, 256, 0, stream>>>(fg, ar_, ai_, br_, bi_, kn, Mp, out);
}